// CTCLoss_24902220382962
// MI455X (gfx1250) — compile-verified
//
#include <hip/hip_runtime.h>
#include <stdint.h>
#include <stddef.h>

// Problem constants from the reference.
#define B_   32
#define T_   800
#define C_   5000
#define L_   100
#define S_   201          // 2*L+1
#define JJ   7            // states per lane: 32*7 = 224 >= 201
#define NEGF (-1e30f)
#define PFD  3            // async gather prefetch distance (ring of 4 LDS buffers)
#define L2D  16           // L2-warm prefetch distance (rows)

// ---------------------------------------------------------------------------
// Kernel 1: one wave32 per utterance. Forward CTC alpha recursion held
// entirely in registers (7 states/lane); cross-state talk via __shfl_up.
// Emission gathers are pipelined PFD steps ahead with CDNA5 async
// global->LDS loads tracked by ASYNCcnt, L2 pre-warmed with global_prefetch.
// ---------------------------------------------------------------------------
__global__ __launch_bounds__(32) void ctc_alpha_kernel(
    const float* __restrict__ nnet,
    const int*   __restrict__ targets,
    const int*   __restrict__ input_lengths,
    const int*   __restrict__ target_lengths,
    float*       __restrict__ ws)
{
  const int b    = blockIdx.x;
  const int lane = threadIdx.x;
  const float* lp = nnet + (size_t)b * (size_t)(T_ * C_);
  const int*   tg = targets + b * L_;
  const int len  = input_lengths[b];
  const int tlen = target_lengths[b];

  __shared__ float ebuf[4][JJ][32];   // async-gather ring buffers (emit[t][s])
  __shared__ float sarr[32 * JJ];     // final alpha spill for score extraction

  // Per-slot class byte-offset and skip-transition flag (s = lane*7 + j).
  // ext[s] = 0 for even s (blank), targets[(s-1)/2] for odd s.
  // skip_ok matches the reference: s==1 has skip_ok=true there but reads the
  // NEG pad, so skip=false here is numerically identical.
  unsigned coff[JJ];
  bool     skip[JJ];
  #pragma unroll
  for (int j = 0; j < JJ; ++j) {
    int s = lane * JJ + j;
    int c = 0; bool sk = false;
    if (s < S_ && (s & 1)) {
      int i = (s - 1) >> 1;
      c  = tg[i];
      sk = (s >= 3) && (c != tg[i - 1]);
    }
    coff[j] = (unsigned)c * 4u;
    skip[j] = sk;
  }

  // LDS byte offset of the ring buffer (low 32 bits of a generic shared
  // pointer == wave-relative LDS address), and the uniform global base.
  const unsigned eb_base = (unsigned)(size_t)(&ebuf[0][0][0]);
  const unsigned long long gbase = (unsigned long long)(size_t)lp;
  const unsigned lane4 = (unsigned)lane * 4u;

  // t = 0 initialization: alpha0[0]=emit[0][0], alpha0[1]=emit[0][1], else NEG.
  float a[JJ];
  #pragma unroll
  for (int j = 0; j < JJ; ++j) a[j] = NEGF;
  if (lane == 0) { a[0] = lp[0]; a[1] = lp[tg[0]]; }

  // Prologue: launch async gathers for emit rows t = 1..PFD (clamped in-bounds).
  for (int tq = 1; tq <= PFD; ++tq) {
    int tp = (tq < len) ? tq : (len - 1);
    unsigned row  = (unsigned)tp * (unsigned)(C_ * 4);
    unsigned boff = (unsigned)(tq & 3) * (JJ * 32u * 4u);
    #pragma unroll
    for (int j = 0; j < JJ; ++j) {
      unsigned ldsa = eb_base + boff + (unsigned)j * (32u * 4u) + lane4;
      unsigned voff = row + coff[j];
      asm volatile("global_load_async_to_lds_b32 %0, %1, %2"
                   :: "v"(ldsa), "v"(voff), "s"(gbase) : "memory");
    }
  }

  // Main serial scan: t = 1 .. len-1 (final alpha is alphas[len-1]).
  for (int t = 1; t < len; ++t) {
    // Warm L2 well ahead of the async gathers (no counter cost).
    {
      int tpf = (t + L2D < len) ? (t + L2D) : (len - 1);
      const char* base = (const char*)lp + (size_t)tpf * (size_t)(C_ * 4);
      #pragma unroll
      for (int j = 0; j < JJ; ++j) __builtin_prefetch(base + coff[j], 0, 1);
    }
    // Issue async gather for emit[t+PFD] into its ring slot.
    {
      int tp = (t + PFD < len) ? (t + PFD) : (len - 1);
      unsigned row  = (unsigned)tp * (unsigned)(C_ * 4);
      unsigned boff = (unsigned)((t + PFD) & 3) * (JJ * 32u * 4u);
      #pragma unroll
      for (int j = 0; j < JJ; ++j) {
        unsigned ldsa = eb_base + boff + (unsigned)j * (32u * 4u) + lane4;
        unsigned voff = row + coff[j];
        asm volatile("global_load_async_to_lds_b32 %0, %1, %2"
                     :: "v"(ldsa), "v"(voff), "s"(gbase) : "memory");
      }
    }
    // Async loads complete in order per wave: once only the PFD newer batches
    // (JJ*PFD ops) are outstanding, emit[t] is fully in LDS.
    asm volatile("s_wait_asynccnt %0" :: "i"(JJ * PFD) : "memory");

    float e[JJ];
    const int bc = t & 3;
    #pragma unroll
    for (int j = 0; j < JJ; ++j) e[j] = ebuf[bc][j][lane];

    // Boundary alphas from the neighboring lane (wave32 shuffle, no barrier).
    float p6 = __shfl_up(a[6], 1);
    float p5 = __shfl_up(a[5], 1);
    if (lane == 0) { p6 = NEGF; p5 = NEGF; }

    float na[JJ];
    #pragma unroll
    for (int j = 0; j < JJ; ++j) {
      float a1  = a[j];
      float a2  = (j == 0) ? p6 : a[j - 1];
      float am2 = (j == 0) ? p5 : ((j == 1) ? p6 : a[j - 2]);
      float a3  = skip[j] ? am2 : NEGF;
      float m   = fmaxf(a1, fmaxf(a2, a3));
      float sum = __expf(a1 - m) + __expf(a2 - m) + __expf(a3 - m);
      na[j] = m + __logf(sum) + e[j];
    }
    #pragma unroll
    for (int j = 0; j < JJ; ++j) a[j] = na[j];
  }
  asm volatile("s_wait_asynccnt 0" ::: "memory");

  // Spill final alpha and extract tot = logaddexp(alpha[2*tl], alpha[2*tl-1]).
  #pragma unroll
  for (int j = 0; j < JJ; ++j) sarr[lane * JJ + j] = a[j];
  __syncthreads();
  if (lane == 0) {
    float x = sarr[2 * tlen];
    float y = sarr[2 * tlen - 1];
    float m = fmaxf(x, y);
    ws[b] = m + __logf(__expf(x - m) + __expf(y - m));
  }
}

// ---------------------------------------------------------------------------
// Kernel 2: single wave32 reduces the 32 per-utterance scores to
// (tot_score, tot_frames, all_frames) as floats.
// ---------------------------------------------------------------------------
__global__ __launch_bounds__(32) void ctc_reduce_kernel(
    const float* __restrict__ ws,
    const int*   __restrict__ input_lengths,
    float*       __restrict__ out)
{
  int lane  = threadIdx.x;
  float tot = ws[lane];
  float len = (float)input_lengths[lane];
  bool finite = tot > (NEGF * 0.5f);
  float s = finite ? tot : 0.0f;
  float f = finite ? len : 0.0f;
  float g = len;
  #pragma unroll
  for (int off = 16; off > 0; off >>= 1) {
    s += __shfl_down(s, off);
    f += __shfl_down(f, off);
    g += __shfl_down(g, off);
  }
  if (lane == 0) { out[0] = s; out[1] = f; out[2] = g; }
}

extern "C" void kernel_launch(void* const* d_in, const int* in_sizes, int n_in,
                              void* d_out, int out_size, void* d_ws, size_t ws_size,
                              hipStream_t stream) {
  const float* nnet    = (const float*)d_in[0];   // (B, T, C) float32
  const int*   targets = (const int*)d_in[1];     // (B, L) int32
  const int*   il      = (const int*)d_in[2];     // (B,) int32
  const int*   tl      = (const int*)d_in[3];     // (B,) int32
  float*       out     = (float*)d_out;           // 3 scalars
  float*       ws      = (float*)d_ws;            // 32 floats of scratch

  ctc_alpha_kernel<<<B_, 32, 0, stream>>>(nnet, targets, il, tl, ws);
  ctc_reduce_kernel<<<1, 32, 0, stream>>>(ws, il, out);
}